// ConcentrationRegressionSNN_86852828660110
// MI455X (gfx1250) — compile-verified
//
#include <hip/hip_runtime.h>

typedef __attribute__((ext_vector_type(2))) float v2f;
typedef __attribute__((ext_vector_type(8))) float v8f;

#define T_STEPS 25
#define BATCH   65536
#define D_IN    57
#define H1      28
#define H2      14
#define BETA    0.9f
#define THR     1.0f

#define WAVES_PER_BLOCK 8
#define ROWS_PER_WAVE   16
#define ROWS_PER_BLOCK  (WAVES_PER_BLOCK * ROWS_PER_WAVE)

__global__ __launch_bounds__(256) void snn_lif_wmma_kernel(
    const float* __restrict__ x,  const float* __restrict__ W1, const float* __restrict__ b1,
    const float* __restrict__ W2, const float* __restrict__ b2,
    const float* __restrict__ W3, const float* __restrict__ b3,
    float* __restrict__ out3, float* __restrict__ out1, float* __restrict__ out2)
{
    __shared__ float lds1[WAVES_PER_BLOCK][ROWS_PER_WAVE * H1]; // spk1 tiles 16x28
    __shared__ float lds2[WAVES_PER_BLOCK][ROWS_PER_WAVE * H2]; // spk2 tiles 16x14

    const int tid  = threadIdx.x;
    const int wave = tid >> 5;
    const int lane = tid & 31;
    const int m    = lane & 15;   // row within tile / N column
    const int h    = lane >> 4;   // lane half (K split for A/B operands)
    const int b0   = blockIdx.x * ROWS_PER_BLOCK + wave * ROWS_PER_WAVE;

    // ---- Pre-stage B operands in registers ----
    // Layer 1: W1^T (57x28), padded K->60 (15 blocks of 4), N->32 (2 tiles)
    v2f B1t[2][15];
    #pragma unroll
    for (int ct = 0; ct < 2; ++ct) {
        const int n = ct * 16 + m;
        #pragma unroll
        for (int kb = 0; kb < 15; ++kb) {
            v2f bt;
            #pragma unroll
            for (int v = 0; v < 2; ++v) {
                const int k = kb * 4 + h * 2 + v;
                bt[v] = (k < D_IN && n < H1) ? W1[n * D_IN + k] : 0.0f;
            }
            B1t[ct][kb] = bt;
        }
    }
    v8f cb1[2];
    #pragma unroll
    for (int ct = 0; ct < 2; ++ct) {
        const int n = ct * 16 + m;
        const float bv = (n < H1) ? b1[n] : 0.0f;
        #pragma unroll
        for (int i = 0; i < 8; ++i) cb1[ct][i] = bv;
    }
    // Layer 2: W2^T (28x14), K exact (7 blocks of 4), N padded 14->16
    v2f B2t[7];
    #pragma unroll
    for (int kb = 0; kb < 7; ++kb) {
        v2f bt;
        #pragma unroll
        for (int v = 0; v < 2; ++v) {
            const int k = kb * 4 + h * 2 + v;
            bt[v] = (m < H2) ? W2[m * H1 + k] : 0.0f;
        }
        B2t[kb] = bt;
    }
    v8f cb2;
    {
        const float bv = (m < H2) ? b2[m] : 0.0f;
        #pragma unroll
        for (int i = 0; i < 8; ++i) cb2[i] = bv;
    }
    float w3r[H2];
    #pragma unroll
    for (int i = 0; i < H2; ++i) w3r[i] = W3[i];
    const float b3v = b3[0];

    // Loop-invariant per-lane-half select mask for the K=56 tail column (branchless, no exec div.)
    const float tailSel = (h == 0) ? 1.0f : 0.0f;

    // ---- Persistent LIF state ----
    v8f mem1[2], spk1[2], mem2, spk2;
    #pragma unroll
    for (int i = 0; i < 8; ++i) {
        mem1[0][i] = 0.0f; mem1[1][i] = 0.0f;
        spk1[0][i] = 0.0f; spk1[1][i] = 0.0f;
        mem2[i] = 0.0f;    spk2[i] = 0.0f;
    }
    float mem3 = 0.0f, spk3 = 0.0f;

    for (int t = 0; t < T_STEPS; ++t) {
        const float* xrow = x + ((size_t)t * BATCH + (size_t)(b0 + m)) * D_IN;

        // A-tile: 16x60 fp32, A-layout (lane half h holds K = 4kb+2h, +1)
        v2f A[15];
        #pragma unroll
        for (int kb = 0; kb < 14; ++kb) {
            const int k = kb * 4 + h * 2;
            v2f a; a[0] = xrow[k]; a[1] = xrow[k + 1];
            A[kb] = a;
        }
        {
            // xrow[56] is a valid address for every lane; select via VALU, not EXEC.
            const float tail = xrow[56];
            v2f a; a[0] = tail * tailSel; a[1] = 0.0f;
            A[14] = a;
        }

        if (t + 1 < T_STEPS)
            __builtin_prefetch(xrow + (size_t)BATCH * D_IN, 0, 1);

        // ---- Layer 1: two 16x16 tiles, 15-deep fp32 WMMA chains ----
        #pragma unroll
        for (int ct = 0; ct < 2; ++ct) {
            v8f cur = cb1[ct];
            #pragma unroll
            for (int kb = 0; kb < 15; ++kb)
                cur = __builtin_amdgcn_wmma_f32_16x16x4_f32(
                    false, A[kb], false, B1t[ct][kb], (short)0, cur, false, false);
            #pragma unroll
            for (int i = 0; i < 8; ++i) {
                const float mv = BETA * mem1[ct][i] + cur[i] - spk1[ct][i] * THR;
                mem1[ct][i] = mv;
                spk1[ct][i] = (mv > THR) ? 1.0f : 0.0f;
            }
        }
        // spk1 C-layout -> LDS row-major 16x28 (vgpr r: row = r + 8*h, col = lane&15)
        #pragma unroll
        for (int r = 0; r < 8; ++r)
            lds1[wave][(r + h * 8) * H1 + m] = spk1[0][r];
        if (m < H1 - 16) {
            #pragma unroll
            for (int r = 0; r < 8; ++r)
                lds1[wave][(r + h * 8) * H1 + 16 + m] = spk1[1][r];
        }
        __syncthreads();

        // spk1 tile is contiguous 448 floats in global: coalesced stream-out
        {
            float* g = out1 + ((size_t)t * BATCH + b0) * H1;
            #pragma unroll
            for (int i = 0; i < 14; ++i) g[i * 32 + lane] = lds1[wave][i * 32 + lane];
        }

        // ---- Layer 2: A from LDS (spk1), 7-deep fp32 WMMA chain ----
        v8f cur2 = cb2;
        #pragma unroll
        for (int kb = 0; kb < 7; ++kb) {
            const int k = kb * 4 + h * 2;
            v2f a; a[0] = lds1[wave][m * H1 + k]; a[1] = lds1[wave][m * H1 + k + 1];
            cur2 = __builtin_amdgcn_wmma_f32_16x16x4_f32(
                false, a, false, B2t[kb], (short)0, cur2, false, false);
        }
        #pragma unroll
        for (int i = 0; i < 8; ++i) {
            const float mv = BETA * mem2[i] + cur2[i] - spk2[i] * THR;
            mem2[i] = mv;
            spk2[i] = (mv > THR) ? 1.0f : 0.0f;
        }
        if (m < H2) {
            #pragma unroll
            for (int r = 0; r < 8; ++r)
                lds2[wave][(r + h * 8) * H2 + m] = spk2[r];
        }
        __syncthreads();

        // spk2 tile contiguous 224 floats: coalesced stream-out
        {
            float* g = out2 + ((size_t)t * BATCH + b0) * H2;
            #pragma unroll
            for (int i = 0; i < 7; ++i) g[i * 32 + lane] = lds2[wave][i * 32 + lane];
        }

        // ---- Layer 3: 14-length dot per row (lanes 16..31 compute duplicates) ----
        float cur3 = b3v;
        #pragma unroll
        for (int k = 0; k < H2; ++k) cur3 += w3r[k] * lds2[wave][m * H2 + k];
        const float mv3 = BETA * mem3 + cur3 - spk3 * THR;
        mem3 = mv3;
        spk3 = (mv3 > THR) ? 1.0f : 0.0f;
        if (lane < 16) out3[(size_t)t * BATCH + b0 + m] = mem3;
    }
}

extern "C" void kernel_launch(void* const* d_in, const int* in_sizes, int n_in,
                              void* d_out, int out_size, void* d_ws, size_t ws_size,
                              hipStream_t stream) {
    (void)in_sizes; (void)n_in; (void)out_size; (void)d_ws; (void)ws_size;
    const float* x  = (const float*)d_in[0];
    const float* W1 = (const float*)d_in[1];
    const float* b1 = (const float*)d_in[2];
    const float* W2 = (const float*)d_in[3];
    const float* b2 = (const float*)d_in[4];
    const float* W3 = (const float*)d_in[5];
    const float* b3 = (const float*)d_in[6];

    float* out3 = (float*)d_out;                                   // mem_rec3 [T,B,1]
    float* out1 = out3 + (size_t)T_STEPS * BATCH;                   // spk_rec1 [T,B,28]
    float* out2 = out1 + (size_t)T_STEPS * BATCH * H1;              // spk_rec2 [T,B,14]

    dim3 grid(BATCH / ROWS_PER_BLOCK);  // 512
    dim3 block(256);                     // 8 wave32
    hipLaunchKernelGGL(snn_lif_wmma_kernel, grid, block, 0, stream,
                       x, W1, b1, W2, b2, W3, b3, out3, out1, out2);
}